// GATEncoder_21431886807832
// MI455X (gfx1250) — compile-verified
//
#include <hip/hip_runtime.h>

// ---------------------------------------------------------------------------
// 2-layer GAT (PyG GATConv semantics) for MI455X / gfx1250, wave32.
//   L1: H=4, C=32 (concat -> 128), ELU ; L2: H=1, C=64 (mean == identity)
// GEMMs use V_WMMA_F32_16X16X4_F32 (fp32 matrix pipe, full precision).
// Edge softmax is done as: atomic-max pass, then one fused pass accumulating
// Sum(exp * h[src]) and Sum(exp) with global_atomic_add_f32 (RMW lands in the
// 192MB L2 since all accumulators fit), then a node-wise divide epilogue.
// ---------------------------------------------------------------------------

typedef float v2f __attribute__((ext_vector_type(2)));
typedef float v8f __attribute__((ext_vector_type(8)));

#define NEG_SLOPE 0.2f

// Order-preserving float <-> uint map so float segment-max can use atomicMax(u32).
__device__ __forceinline__ unsigned f2ord(float f) {
  unsigned u = __float_as_uint(f);
  return (u & 0x80000000u) ? ~u : (u | 0x80000000u);
}
__device__ __forceinline__ float ord2f(unsigned u) {
  return __uint_as_float((u & 0x80000000u) ? (u & 0x7FFFFFFFu) : ~u);
}
#define ORD_NEG_INF 0x007FFFFFu  // f2ord(-inf)

// ---------------------------------------------------------------------------
__global__ void fill_u32(unsigned* __restrict__ p, unsigned v, long long n) {
  long long t = (long long)blockIdx.x * blockDim.x + threadIdx.x;
  if (t < n) p[t] = v;
}

// ---------------------------------------------------------------------------
// C[M,N] = A[M,K] @ B[K,N], fp32, one wave per 16x16 tile via WMMA f32 16x16x4.
// A 16x4 layout: lanes 0-15 hold K={k0,k0+1}, lanes 16-31 hold K={k0+2,k0+3},
// row = lane%16.  B 4x16: mirrored (col = lane%16).  C/D: vgpr r holds row
// r + 8*(lane>=16), col = lane%16.
__global__ void gemm_wmma_f32(const float* __restrict__ A,
                              const float* __restrict__ B,
                              float* __restrict__ C, int M, int K, int N) {
  const int lane = threadIdx.x & 31;
  const int wave = blockIdx.x * (blockDim.x >> 5) + (threadIdx.x >> 5);
  const int ntiles = N >> 4;
  const int mt = wave / ntiles;
  const int nt = wave - mt * ntiles;
  if (mt * 16 >= M) return;  // wave-uniform: EXEC stays all-ones for WMMA

  const int r16 = lane & 15;
  const int hi = lane >> 4;  // 0: lanes 0-15, 1: lanes 16-31
  const int koff = hi * 2;
  const long long arow = (long long)(mt * 16 + r16) * K;
  const int bcol = nt * 16 + r16;

  v8f acc = {};
  for (int k0 = 0; k0 < K; k0 += 4) {
    v2f a, b;
    a[0] = A[arow + k0 + koff];
    a[1] = A[arow + k0 + koff + 1];
    b[0] = B[(long long)(k0 + koff) * N + bcol];
    b[1] = B[(long long)(k0 + koff + 1) * N + bcol];
    acc = __builtin_amdgcn_wmma_f32_16x16x4_f32(
        false, a, false, b, (short)0, acc, false, false);
  }
  const long long crow = (long long)(mt * 16 + hi * 8);
#pragma unroll
  for (int r = 0; r < 8; ++r) C[(crow + r) * N + bcol] = acc[r];
}

// ---------------------------------------------------------------------------
// als[n,h] = dot(h[n, h*C:(h+1)*C], a_src[h]);  ald likewise.
__global__ void attn_logits(const float* __restrict__ h,
                            const float* __restrict__ a_s,
                            const float* __restrict__ a_d,
                            float* __restrict__ als, float* __restrict__ ald,
                            int N, int H, int C) {
  int t = blockIdx.x * blockDim.x + threadIdx.x;
  if (t >= N * H) return;
  int n = t / H, hh = t - (t / H) * H;
  const float* hp = h + (long long)n * H * C + hh * C;
  const float* as = a_s + hh * C;
  const float* ad = a_d + hh * C;
  float s = 0.f, d = 0.f;
  for (int c = 0; c < C; ++c) {
    float v = hp[c];
    s += v * as[c];
    d += v * ad[c];
  }
  als[t] = s;
  ald[t] = d;
}

// ---------------------------------------------------------------------------
// Segment max over incoming edges: one thread per (edge, head).
__global__ void edge_max(const int* __restrict__ ei, int E, int N, int hshift,
                         const float* __restrict__ als,
                         const float* __restrict__ ald,
                         unsigned* __restrict__ m) {
  long long t = (long long)blockIdx.x * blockDim.x + threadIdx.x;
  const long long tot = ((long long)E + N) << hshift;
  if (t >= tot) return;
  const int H = 1 << hshift;
  const int hh = (int)(t & (H - 1));
  const long long e = t >> hshift;
  int src, dst;
  if (e < E) { src = ei[e]; dst = ei[(long long)E + e]; }
  else       { src = dst = (int)(e - E); }
  float lg = als[src * H + hh] + ald[dst * H + hh];
  lg = lg > 0.f ? lg : NEG_SLOPE * lg;
  atomicMax(&m[dst * H + hh], f2ord(lg));
}

// ---------------------------------------------------------------------------
// Layer-1 fused exp + weighted scatter. One wave per edge; lane l owns
// channel h*32+l for each head h in 0..3 (coalesced 128B per head).
__global__ void edge_accum_l1(const int* __restrict__ ei, int E, int N,
                              const float* __restrict__ h,
                              const float* __restrict__ als,
                              const float* __restrict__ ald,
                              const unsigned* __restrict__ m,
                              float* __restrict__ denom,
                              float* __restrict__ acc) {
  const int lane = threadIdx.x & 31;
  long long e = (long long)blockIdx.x * (blockDim.x >> 5) + (threadIdx.x >> 5);
  if (e >= (long long)E + N) return;
  int src, dst;
  if (e < E) { src = ei[e]; dst = ei[(long long)E + e]; }
  else       { src = dst = (int)(e - E); }
  const long long hs = (long long)src * 128;
  const long long hd = (long long)dst * 128;
#pragma unroll
  for (int hh = 0; hh < 4; ++hh) {
    float lg = als[src * 4 + hh] + ald[dst * 4 + hh];
    lg = lg > 0.f ? lg : NEG_SLOPE * lg;
    float w = __expf(lg - ord2f(m[dst * 4 + hh]));
    if (lane == hh) atomicAdd(&denom[dst * 4 + hh], w);
    const int ch = hh * 32 + lane;
    atomicAdd(&acc[hd + ch], w * h[hs + ch]);
  }
}

// ---------------------------------------------------------------------------
// Layer-2 fused exp + weighted scatter (H=1, F=64). Lane owns ch l and l+32.
__global__ void edge_accum_l2(const int* __restrict__ ei, int E, int N,
                              const float* __restrict__ h,
                              const float* __restrict__ als,
                              const float* __restrict__ ald,
                              const unsigned* __restrict__ m,
                              float* __restrict__ denom,
                              float* __restrict__ out) {
  const int lane = threadIdx.x & 31;
  long long e = (long long)blockIdx.x * (blockDim.x >> 5) + (threadIdx.x >> 5);
  if (e >= (long long)E + N) return;
  int src, dst;
  if (e < E) { src = ei[e]; dst = ei[(long long)E + e]; }
  else       { src = dst = (int)(e - E); }
  float lg = als[src] + ald[dst];
  lg = lg > 0.f ? lg : NEG_SLOPE * lg;
  float w = __expf(lg - ord2f(m[dst]));
  if (lane == 0) atomicAdd(&denom[dst], w);
  const long long hs = (long long)src * 64;
  const long long hd = (long long)dst * 64;
  atomicAdd(&out[hd + lane], w * h[hs + lane]);
  atomicAdd(&out[hd + lane + 32], w * h[hs + lane + 32]);
}

// ---------------------------------------------------------------------------
// out = ELU(acc/denom + b1)  (layer-1 epilogue, 128 channels, 4 heads)
__global__ void epilogue1(const float* __restrict__ acc,
                          const float* __restrict__ denom,
                          const float* __restrict__ b, float* __restrict__ out,
                          long long total) {
  long long t = (long long)blockIdx.x * blockDim.x + threadIdx.x;
  if (t >= total) return;
  const int ch = (int)(t & 127);
  const long long n = t >> 7;
  float v = acc[t] / denom[n * 4 + (ch >> 5)] + b[ch];
  out[t] = v > 0.f ? v : (__expf(v) - 1.f);
}

// out = out/denom + b2   (layer-2 epilogue, in place, 64 channels)
__global__ void finalize(float* __restrict__ out,
                         const float* __restrict__ denom,
                         const float* __restrict__ b, long long total) {
  long long t = (long long)blockIdx.x * blockDim.x + threadIdx.x;
  if (t >= total) return;
  const int c = (int)(t & 63);
  const long long n = t >> 6;
  out[t] = out[t] / denom[n] + b[c];
}

// ---------------------------------------------------------------------------
extern "C" void kernel_launch(void* const* d_in, const int* in_sizes, int n_in,
                              void* d_out, int out_size, void* d_ws,
                              size_t ws_size, hipStream_t stream) {
  const float* x = (const float*)d_in[0];
  const int* ei = (const int*)d_in[1];
  const float* W1 = (const float*)d_in[2];
  const float* a1s = (const float*)d_in[3];
  const float* a1d = (const float*)d_in[4];
  const float* b1 = (const float*)d_in[5];
  const float* W2 = (const float*)d_in[6];
  const float* a2s = (const float*)d_in[7];
  const float* a2d = (const float*)d_in[8];
  const float* b2 = (const float*)d_in[9];
  float* out = (float*)d_out;

  const int N = in_sizes[0] / 128;
  const int E = in_sizes[1] / 2;
  const long long EN = (long long)E + N;

  // Workspace carving (floats). Total ~= 112 MB for N=100k.
  float* ws = (float*)d_ws;
  long long off = 0;
  float* h1 = ws + off;    off += (long long)N * 128;  // layer-1 features / ELU out
  float* acc1 = ws + off;  off += (long long)N * 128;  // L1 accumulator, reused as h2
  float* als1 = ws + off;  off += (long long)N * 4;
  float* ald1 = ws + off;  off += (long long)N * 4;
  unsigned* m1 = (unsigned*)(ws + off); off += (long long)N * 4;
  float* den1 = ws + off;  off += (long long)N * 4;
  float* als2 = ws + off;  off += N;
  float* ald2 = ws + off;  off += N;
  unsigned* m2 = (unsigned*)(ws + off); off += N;
  float* den2 = ws + off;  off += N;
  float* h2 = acc1;  // acc1 is dead after epilogue1; reuse for layer-2 features

  const int TB = 256;  // 8 waves per block (wave32)

  // ---- Layer 1 ----
  {
    const long long waves = (long long)((N + 15) / 16) * (128 / 16);
    const int blocks = (int)((waves + 7) / 8);
    gemm_wmma_f32<<<blocks, TB, 0, stream>>>(x, W1, h1, N, 128, 128);
  }
  attn_logits<<<(N * 4 + TB - 1) / TB, TB, 0, stream>>>(h1, a1s, a1d, als1,
                                                        ald1, N, 4, 32);
  fill_u32<<<(int)(((long long)N * 4 + TB - 1) / TB), TB, 0, stream>>>(
      m1, ORD_NEG_INF, (long long)N * 4);
  fill_u32<<<(int)(((long long)N * 4 + TB - 1) / TB), TB, 0, stream>>>(
      (unsigned*)den1, 0u, (long long)N * 4);
  fill_u32<<<(int)(((long long)N * 128 + TB - 1) / TB), TB, 0, stream>>>(
      (unsigned*)acc1, 0u, (long long)N * 128);
  fill_u32<<<(N + TB - 1) / TB, TB, 0, stream>>>(m2, ORD_NEG_INF, N);
  fill_u32<<<(N + TB - 1) / TB, TB, 0, stream>>>((unsigned*)den2, 0u, N);
  fill_u32<<<(int)(((long long)N * 64 + TB - 1) / TB), TB, 0, stream>>>(
      (unsigned*)out, 0u, (long long)N * 64);

  edge_max<<<(int)((EN * 4 + TB - 1) / TB), TB, 0, stream>>>(ei, E, N, 2, als1,
                                                             ald1, m1);
  edge_accum_l1<<<(int)((EN + 7) / 8), TB, 0, stream>>>(ei, E, N, h1, als1,
                                                        ald1, m1, den1, acc1);
  epilogue1<<<(int)(((long long)N * 128 + TB - 1) / TB), TB, 0, stream>>>(
      acc1, den1, b1, h1, (long long)N * 128);

  // ---- Layer 2 ----
  {
    const long long waves = (long long)((N + 15) / 16) * (64 / 16);
    const int blocks = (int)((waves + 7) / 8);
    gemm_wmma_f32<<<blocks, TB, 0, stream>>>(h1, W2, h2, N, 128, 64);
  }
  attn_logits<<<(N + TB - 1) / TB, TB, 0, stream>>>(h2, a2s, a2d, als2, ald2,
                                                    N, 1, 64);
  edge_max<<<(int)((EN + TB - 1) / TB), TB, 0, stream>>>(ei, E, N, 0, als2,
                                                         ald2, m2);
  edge_accum_l2<<<(int)((EN + 7) / 8), TB, 0, stream>>>(ei, E, N, h2, als2,
                                                        ald2, m2, den2, out);
  finalize<<<(int)(((long long)N * 64 + TB - 1) / TB), TB, 0, stream>>>(
      out, den2, b2, (long long)N * 64);
}